// ParabolicPool1DFast_79078937854425
// MI455X (gfx1250) — compile-verified
//
#include <hip/hip_runtime.h>
#include <cstdint>

// Problem constants (match reference)
#define NB    32
#define NC    64
#define NL    4096
#define OUTL  2048      // NL / STRIDE
#define TABN  8192      // 2 * NL : only out[0,0,:] and out[0,1,:] are ever read

// ---- CDNA5 async global->LDS helpers (ASYNCcnt-tracked, gfx1250) ------------
// Low 32 bits of a flat pointer into LDS == wave-relative LDS byte offset
// (CDNA5 ISA: in the LDS aperture the low 32 bits are the LDS address).
__device__ __forceinline__ uint32_t lds_byte_off(const void* p) {
  return (uint32_t)(uintptr_t)p;
}

// global_load_async_to_lds_b128 vdst(lds byte addr), vaddr(64b global), off
__device__ __forceinline__ void async_copy_b128(uint32_t lds_byte, const void* gaddr) {
  asm volatile("global_load_async_to_lds_b128 %0, %1, off"
               :: "v"(lds_byte), "v"(gaddr)
               : "memory");
}

__device__ __forceinline__ void wait_async_zero() {
  asm volatile("s_wait_asynccnt 0" ::: "memory");
}

// ---- Kernel A: parabolic (max-plus) dilation of the two live channels -------
// table[c*NL + l] = max_{d=-3..3} f[0,c,l+d] + h_c[d],  h_c[d] = -(d*d)/(4*t_c)
__global__ __launch_bounds__(256)
void pp_build_table(const float* __restrict__ f,
                    const float* __restrict__ t,
                    float* __restrict__ table) {
  __shared__ float row[NL];                 // 16 KB
  const int c   = blockIdx.x;               // 0 or 1
  const int tid = threadIdx.x;
  const float* __restrict__ frow = f + (size_t)c * NL;   // batch 0, channel c

  // Async-stage the 16 KB row into LDS: 256 threads x 4 x 16B
  const uint32_t lds_base = lds_byte_off(row);
  #pragma unroll
  for (int v = 0; v < 4; ++v) {
    const int e4 = v * 256 + tid;           // float4 index 0..1023
    async_copy_b128(lds_base + (uint32_t)e4 * 16u, frow + e4 * 4);
  }
  wait_async_zero();
  __syncthreads();

  // Faithful to reference float arithmetic: h = -(z^2) / (4*t); z^2 in {0,1,4,9}
  const float q  = 4.0f * t[c];
  const float h1 = -1.0f / q;
  const float h4 = -4.0f / q;
  const float h9 = -9.0f / q;

  #pragma unroll
  for (int v = 0; v < 16; ++v) {
    const int l = v * 256 + tid;            // 0..4095, coalesced stores
    float m = row[l];                       // d=0 tap (h=-0.0f: x + -0.0 == x)
    if (l >= 1)      m = fmaxf(m, row[l - 1] + h1);
    if (l + 1 < NL)  m = fmaxf(m, row[l + 1] + h1);
    if (l >= 2)      m = fmaxf(m, row[l - 2] + h4);
    if (l + 2 < NL)  m = fmaxf(m, row[l + 2] + h4);
    if (l >= 3)      m = fmaxf(m, row[l - 3] + h9);
    if (l + 3 < NL)  m = fmaxf(m, row[l + 3] + h9);   // borders: -inf pad == skip
    table[c * NL + l] = m;
  }
}

// ---- Kernel B: strided-gather broadcast (the as_strided reproduction) -------
// d_out[((b*NC)+c)*OUTL + i] = table[b + c + 2*i]
__global__ __launch_bounds__(256)
void pp_scatter(const float* __restrict__ table,
                float4* __restrict__ out4, int n4) {
  __shared__ float tab[TABN];               // 32 KB (of 320 KB/WGP)
  const int tid = threadIdx.x;

  // Async-stage the whole gather table into LDS: 256 threads x 8 x 16B
  const uint32_t lds_base = lds_byte_off(tab);
  #pragma unroll
  for (int v = 0; v < 8; ++v) {
    const int e4 = v * 256 + tid;           // 0..2047
    async_copy_b128(lds_base + (uint32_t)e4 * 16u, table + e4 * 4);
  }
  wait_async_zero();
  __syncthreads();

  const int gid = blockIdx.x * 256 + tid;   // float4 output index
  if (gid < n4) {
    const int o  = gid * 4;                 // flat output element index
    const int i  = o & (OUTL - 1);
    const int bc = o >> 11;                 // o / OUTL
    const int c  = bc & (NC - 1);
    const int b  = bc >> 6;                 // bc / NC
    const int k  = b + c + 2 * i;           // <= 4188 < TABN
    float4 r;
    r.x = tab[k];
    r.y = tab[k + 2];
    r.z = tab[k + 4];
    r.w = tab[k + 6];
    out4[gid] = r;                          // coalesced 16B stores -> HBM-bound
  }
}

extern "C" void kernel_launch(void* const* d_in, const int* in_sizes, int n_in,
                              void* d_out, int out_size, void* d_ws, size_t ws_size,
                              hipStream_t stream) {
  const float* f = (const float*)d_in[0];   // [32,64,4096] f32
  const float* t = (const float*)d_in[1];   // [64] f32
  float* table = (float*)d_ws;              // needs 32 KB scratch

  pp_build_table<<<dim3(2), dim3(256), 0, stream>>>(f, t, table);

  const int n4 = out_size / 4;              // 1,048,576 float4 stores
  pp_scatter<<<dim3((n4 + 255) / 256), dim3(256), 0, stream>>>(
      table, (float4*)d_out, n4);
}